// StateSpaceLayer_23854248362680
// MI455X (gfx1250) — compile-verified
//
#include <hip/hip_runtime.h>
#include <math.h>

// ---- problem constants (match reference) ----
#define BSZ   2
#define SEQ   2048
#define DM    512           // d_model
#define DS    64            // d_state
#define DTR   32            // dt_rank
#define DCONV 4
#define XPLD  (DTR + 2*DS)  // 160, x_proj output width
#define NROWS (BSZ*SEQ)     // 4096 flattened (b,l) rows
#define CH    32            // scan chunks
#define LC    (SEQ/CH)      // 64 timesteps per chunk

typedef float v2f __attribute__((ext_vector_type(2)));
typedef float v8f __attribute__((ext_vector_type(8)));
typedef unsigned int u32x4 __attribute__((ext_vector_type(4)));
typedef int i32x4 __attribute__((ext_vector_type(4)));
typedef int i32x8 __attribute__((ext_vector_type(8)));

#if defined(__has_builtin)
#if __has_builtin(__builtin_amdgcn_tensor_load_to_lds) && __has_builtin(__builtin_amdgcn_s_wait_tensorcnt)
#define USE_TDM 1
#endif
#endif
#ifndef USE_TDM
#define USE_TDM 0
#endif

__device__ __forceinline__ float softplus_f(float x) {
    return (x > 20.0f) ? x : log1pf(expf(x));
}

// =====================================================================
// f32 WMMA GEMM:  C[M,N] = A[M,K](lda) * B[K,N](ldb) + bias[N]
// Block = 256 threads = 8 waves = 8 M-tiles x 1 N-tile.  The K x 16
// B panel (<=32KB) is staged into LDS once per block by the Tensor
// Data Mover (one tensor_load_to_lds issued by wave 0, D# built in
// SGPRs, completion via s_wait_tensorcnt + barrier), then each wave
// runs the K/4 chain of V_WMMA_F32_16X16X4_F32 with A fragments from
// global (b64 per lane) and B fragments from LDS (conflict-free).
// act: 0 = none, 1 = softplus (dt_proj).
// =====================================================================
__global__ void gemm16_wmma_f32(const float* __restrict__ A, int lda,
                                const float* __restrict__ Bm, int ldb,
                                const float* __restrict__ bias,
                                float* __restrict__ Cm, int ldc,
                                int Nt, int K, int act) {
    __shared__ float bpanel[512 * 16];          // supports K <= 512
    const int lane = threadIdx.x & 31;
    const int wv   = threadIdx.x >> 5;          // wave in block: 0..7
    const int tn   = blockIdx.x % Nt;           // shared N tile
    const int tm   = (blockIdx.x / Nt) * 8 + wv;
    const int coloff = tn * 16;

#if USE_TDM
    if (wv == 0) {
        // ---- Tensor DMA descriptor (D#): 2D tile, K rows x 16 floats ----
        const unsigned long long ga = (unsigned long long)(uintptr_t)(Bm + coloff);
        const unsigned int ldsa = (unsigned int)(uintptr_t)(&bpanel[0]);
        u32x4 g0;
        g0[0] = 1u;                                               // count=1 (valid user D#)
        g0[1] = ldsa;                                             // lds_addr (bytes)
        g0[2] = (unsigned int)(ga & 0xFFFFFFFFu);                 // global_addr[31:0]
        g0[3] = (unsigned int)((ga >> 32) & 0x01FFFFFFu)          // global_addr[56:32]
              | 0x80000000u;                                      // type = 2 ("image")
        i32x8 g1;
        g1[0] = 0x00020000;                                       // data_size = 4B
        g1[1] = (int)(16u << 16);                                 // tensor_dim0 = 16
        g1[2] = (int)((unsigned)K << 16);                         // tensor_dim1[15:0] = K
        g1[3] = (int)((((unsigned)K >> 16) & 0xFFFFu))            // tensor_dim1[31:16]
              | (16 << 16);                                       // tile_dim0 = 16
        g1[4] = (int)((unsigned)K & 0xFFFFu);                     // tile_dim1 = K
        g1[5] = ldb;                                              // tensor_dim0_stride
        g1[6] = 0;
        g1[7] = 0;
        i32x4 z4 = {};
        i32x8 z8 = {};
#if __clang_major__ >= 23
        __builtin_amdgcn_tensor_load_to_lds(g0, g1, z4, z4, z8, 0);
#else
        __builtin_amdgcn_tensor_load_to_lds(g0, g1, z4, z4, 0);
#endif
        __builtin_amdgcn_s_wait_tensorcnt(0);
    }
    __syncthreads();
#else
    for (int i = threadIdx.x; i < K * 16; i += 256)
        bpanel[i] = Bm[(size_t)(i >> 4) * ldb + coloff + (i & 15)];
    __syncthreads();
#endif

    const int mrow = tm * 16 + (lane & 15);     // A row this lane reads
    const int ncol = coloff + (lane & 15);      // C column this lane owns
    const int kofs = (lane >> 4) << 1;          // K sub-pair per half-wave

    v8f acc = {};
#pragma unroll 4
    for (int k = 0; k < K; k += 4) {
        v2f a, b;
        a.x = A[(size_t)mrow * lda + (k + kofs)];
        a.y = A[(size_t)mrow * lda + (k + kofs + 1)];
        b.x = bpanel[(k + kofs) * 16 + (lane & 15)];
        b.y = bpanel[(k + kofs + 1) * 16 + (lane & 15)];
        acc = __builtin_amdgcn_wmma_f32_16x16x4_f32(
                  /*neg_a=*/false, a, /*neg_b=*/false, b,
                  /*c_mod=*/(short)0, acc, /*reuse_a=*/false, /*reuse_b=*/false);
    }

    const float bv = bias[ncol];
    const int rbase = tm * 16 + ((lane >> 4) << 3);
#pragma unroll
    for (int r = 0; r < 8; ++r) {
        float v = acc[r] + bv;
        if (act == 1) v = softplus_f(v);
        Cm[(size_t)(rbase + r) * ldc + ncol] = v;
    }
}

// =====================================================================
// Depthwise causal conv1d, K=4:  y[b,t,d] = cb[d] + sum_k w[d,k]*x[b,t-3+k,d]
// x is the first-half (xc) slice of xz laid out [B,L,2*DM].
// =====================================================================
__global__ void dwconv_k4(const float* __restrict__ xz,
                          const float* __restrict__ cw,
                          const float* __restrict__ cb,
                          float* __restrict__ xc) {
    const int idx = blockIdx.x * blockDim.x + threadIdx.x;
    if (idx >= BSZ * SEQ * DM) return;
    const int d = idx % DM;
    const int t = (idx / DM) % SEQ;
    const int b = idx / (DM * SEQ);
    float acc = cb[d];
#pragma unroll
    for (int k = 0; k < DCONV; ++k) {
        const int tt = t - (DCONV - 1) + k;
        if (tt >= 0)
            acc += cw[d * DCONV + k] * xz[(size_t)(b * SEQ + tt) * (2 * DM) + d];
    }
    xc[idx] = acc;   // idx == (b*SEQ+t)*DM + d  (contiguous [B,L,DM])
}

// =====================================================================
// Scan phase 1: one wave per (b,d,chunk).  Lane owns states n=lane and
// n=lane+32.  Local scan from h=0; record end-state and a-product.
// =====================================================================
__global__ void scan_phase1(const float* __restrict__ dt,
                            const float* __restrict__ xc,
                            const float* __restrict__ xp,
                            const float* __restrict__ A_log,
                            const float* __restrict__ Bpar,
                            float* __restrict__ hend,
                            float* __restrict__ Pend) {
    const int wid  = (blockIdx.x * blockDim.x + threadIdx.x) >> 5;
    const int lane = threadIdx.x & 31;
    const int c = wid % CH;
    const int d = (wid / CH) % DM;
    const int b = wid / (CH * DM);
    const int n0 = lane, n1 = lane + 32;

    const float A0 = -expf(A_log[d * DS + n0]);
    const float A1 = -expf(A_log[d * DS + n1]);
    const float Bw0 = Bpar[d * DS + n0];
    const float Bw1 = Bpar[d * DS + n1];

    float h0 = 0.f, h1 = 0.f, P0 = 1.f, P1 = 1.f;
    const int t0 = c * LC;
    for (int t = t0; t < t0 + LC; ++t) {
        const size_t row = (size_t)(b * SEQ + t);
        const float dtv = dt[row * DM + d];
        const float u   = xc[row * DM + d];
        const float bp0 = xp[row * XPLD + DTR + n0];
        const float bp1 = xp[row * XPLD + DTR + n1];
        const float a0 = expf(dtv * A0);
        const float a1 = expf(dtv * A1);
        const float du = dtv * u;
        h0 = fmaf(a0, h0, Bw0 * bp0 * du);
        h1 = fmaf(a1, h1, Bw1 * bp1 * du);
        P0 *= a0; P1 *= a1;
    }
    const size_t base = (((size_t)(b * DM + d)) * CH + c) * DS;
    hend[base + n0] = h0;  hend[base + n1] = h1;
    Pend[base + n0] = P0;  Pend[base + n1] = P1;
}

// =====================================================================
// Scan phase 2: chain chunk boundary states.  Thread per (b,d,n), 32-step
// serial loop over chunks.  hinit[c] = state entering chunk c.
// =====================================================================
__global__ void scan_phase2(const float* __restrict__ hend,
                            const float* __restrict__ Pend,
                            float* __restrict__ hinit) {
    const int idx = blockIdx.x * blockDim.x + threadIdx.x;
    if (idx >= BSZ * DM * DS) return;
    const int n = idx % DS;
    const int d = (idx / DS) % DM;
    const int b = idx / (DS * DM);
    const size_t base = ((size_t)(b * DM + d)) * CH;
    float h = 0.f;
    for (int c = 0; c < CH; ++c) {
        hinit[(base + c) * DS + n] = h;
        h = Pend[(base + c) * DS + n] * h + hend[(base + c) * DS + n];
    }
}

// =====================================================================
// Scan phase 3: replay each chunk from corrected init state; per step
// reduce y = sum_n Cp*C*h over the wave (N=64, 2 per lane) and fuse
// skip connection + SiLU gate.
// =====================================================================
__global__ void scan_phase3(const float* __restrict__ dt,
                            const float* __restrict__ xc,
                            const float* __restrict__ xp,
                            const float* __restrict__ xz,
                            const float* __restrict__ A_log,
                            const float* __restrict__ Bpar,
                            const float* __restrict__ Cpar,
                            const float* __restrict__ Dpar,
                            const float* __restrict__ hinit,
                            float* __restrict__ out) {
    const int wid  = (blockIdx.x * blockDim.x + threadIdx.x) >> 5;
    const int lane = threadIdx.x & 31;
    const int c = wid % CH;
    const int d = (wid / CH) % DM;
    const int b = wid / (CH * DM);
    const int n0 = lane, n1 = lane + 32;

    const float A0 = -expf(A_log[d * DS + n0]);
    const float A1 = -expf(A_log[d * DS + n1]);
    const float Bw0 = Bpar[d * DS + n0];
    const float Bw1 = Bpar[d * DS + n1];
    const float Cw0 = Cpar[d * DS + n0];
    const float Cw1 = Cpar[d * DS + n1];
    const float Dp  = Dpar[d];

    const size_t hb = (((size_t)(b * DM + d)) * CH + c) * DS;
    float h0 = hinit[hb + n0];
    float h1 = hinit[hb + n1];

    const int t0 = c * LC;
    for (int t = t0; t < t0 + LC; ++t) {
        const size_t row = (size_t)(b * SEQ + t);
        const float dtv = dt[row * DM + d];
        const float u   = xc[row * DM + d];
        const float bp0 = xp[row * XPLD + DTR + n0];
        const float bp1 = xp[row * XPLD + DTR + n1];
        const float cp0 = xp[row * XPLD + DTR + DS + n0];
        const float cp1 = xp[row * XPLD + DTR + DS + n1];
        const float a0 = expf(dtv * A0);
        const float a1 = expf(dtv * A1);
        const float du = dtv * u;
        h0 = fmaf(a0, h0, Bw0 * bp0 * du);
        h1 = fmaf(a1, h1, Bw1 * bp1 * du);
        float y = Cw0 * cp0 * h0 + Cw1 * cp1 * h1;
#pragma unroll
        for (int off = 16; off > 0; off >>= 1)
            y += __shfl_xor(y, off, 32);        // wave32 butterfly reduce over n
        if (lane == 0) {
            const float z = xz[row * (2 * DM) + DM + d];
            const float sig = 1.f / (1.f + expf(-z));
            out[row * DM + d] = (y + Dp * u) * (z * sig);
        }
    }
}

// =====================================================================
extern "C" void kernel_launch(void* const* d_in, const int* in_sizes, int n_in,
                              void* d_out, int out_size, void* d_ws, size_t ws_size,
                              hipStream_t stream) {
    const float* x      = (const float*)d_in[0];
    const float* in_w   = (const float*)d_in[1];   // [512,1024]
    const float* in_b   = (const float*)d_in[2];
    const float* conv_w = (const float*)d_in[3];   // [512,1,4]
    const float* conv_b = (const float*)d_in[4];
    const float* xpw    = (const float*)d_in[5];   // [512,160]
    const float* xpb    = (const float*)d_in[6];
    const float* dtw    = (const float*)d_in[7];   // [32,512]
    const float* dtb    = (const float*)d_in[8];
    const float* A_log  = (const float*)d_in[9];
    const float* Bpar   = (const float*)d_in[10];
    const float* Cpar   = (const float*)d_in[11];
    const float* Dpar   = (const float*)d_in[12];
    float* out = (float*)d_out;

    float* ws    = (float*)d_ws;
    float* xz    = ws;                                   // 4096*1024
    float* xcv   = xz   + (size_t)NROWS * 2 * DM;        // 4096*512
    float* xp    = xcv  + (size_t)NROWS * DM;            // 4096*160
    float* dt    = xp   + (size_t)NROWS * XPLD;          // 4096*512
    float* hend  = dt   + (size_t)NROWS * DM;            // 2*512*32*64
    float* Pend  = hend + (size_t)BSZ * DM * CH * DS;
    float* hinit = Pend + (size_t)BSZ * DM * CH * DS;

    const int MtBlocks = (NROWS / 16) / 8;               // 32 blocks of 8 M-tiles

    // 1) in_proj: xz = x @ in_proj_w + b        [4096 x 1024], K=512
    gemm16_wmma_f32<<<MtBlocks * ((2 * DM) / 16), 256, 0, stream>>>(
        x, DM, in_w, 2 * DM, in_b, xz, 2 * DM, (2 * DM) / 16, DM, /*act=*/0);

    // 2) depthwise causal conv on xc = xz[:, :512]
    dwconv_k4<<<(BSZ * SEQ * DM) / 256, 256, 0, stream>>>(xz, conv_w, conv_b, xcv);

    // 3) x_proj: xp = xcv @ x_proj_w + b        [4096 x 160], K=512
    gemm16_wmma_f32<<<MtBlocks * (XPLD / 16), 256, 0, stream>>>(
        xcv, DM, xpw, XPLD, xpb, xp, XPLD, XPLD / 16, DM, /*act=*/0);

    // 4) dt = softplus(xp[:, :32] @ dt_proj_w + b)   [4096 x 512], K=32
    gemm16_wmma_f32<<<MtBlocks * (DM / 16), 256, 0, stream>>>(
        xp, XPLD, dtw, DM, dtb, dt, DM, DM / 16, DTR, /*act=*/1);

    // 5) chunked parallel scan + fused output
    scan_phase1<<<(BSZ * DM * CH) / 8, 256, 0, stream>>>(dt, xcv, xp, A_log, Bpar, hend, Pend);
    scan_phase2<<<(BSZ * DM * DS) / 256, 256, 0, stream>>>(hend, Pend, hinit);
    scan_phase3<<<(BSZ * DM * CH) / 8, 256, 0, stream>>>(dt, xcv, xp, xz, A_log, Bpar,
                                                         Cpar, Dpar, hinit, out);
}